// SPP_40467181863103
// MI455X (gfx1250) — compile-verified
//
#include <hip/hip_runtime.h>
#include <cstdint>
#include <cstddef>

// ---------------------------------------------------------------------------
// SPP (x, maxpool5, maxpool9, maxpool13 concat) for NHWC (16,64,64,512) f32.
// Memory-bound: min HBM traffic = 134MB R + 537MB W ~= 29us @ 23.3TB/s.
// SPPF composition pool13 = pool5(pool5(pool5)), separable h/v max-5,
// register sliding scans (horizontal) + register rings (vertical) over LDS
// tiles, gfx1250 async global->LDS loads, non-temporal output stores so the
// input stays resident in the 192MB L2 (halo re-reads never touch HBM).
// LDS uses an XOR bank swizzle (no padding) -> exactly 163,840B per block,
// two workgroups per WGP (16 wave32) for latency hiding.
// ---------------------------------------------------------------------------

typedef float v4f __attribute__((ext_vector_type(4)));

#ifndef SPP_USE_ASYNC
#define SPP_USE_ASYNC 1
#endif

namespace {
constexpr int Hn = 64, Wn = 64, Cn = 512, OCn = 2048;
constexpr int TH = 8;            // output rows per block tile
constexpr int CC = 16;           // channels per block
constexpr int C4 = CC / 4;       // float4 chunks per pixel per block (4)
constexpr int RRAW = TH + 12;    // 20 input rows (+/-6 halo for 13x13)
constexpr int ROWF4 = Wn * C4;   // 256 float4 per row, no pad (XOR swizzle)
constexpr int BUF = RRAW * ROWF4;  // 5120 float4 per buffer
constexpr int NTHREADS = 256;
}  // namespace

// Bank swizzle: bijection on w in [0,64). XORs low 2 bits of w with the
// 16-wide run id so the four scan runs hit all four 16-bank quarters.
__device__ __forceinline__ int wslot(int w) {
  return (w ^ ((w >> 4) & 3)) * C4;
}

__device__ __forceinline__ v4f vmax4(v4f a, v4f b) {
  v4f r;
  r.x = fmaxf(a.x, b.x);
  r.y = fmaxf(a.y, b.y);
  r.z = fmaxf(a.z, b.z);
  r.w = fmaxf(a.w, b.w);
  return r;
}

__device__ __forceinline__ v4f neg_inf4() {
  const float ni = -__builtin_inff();
  v4f r = {ni, ni, ni, ni};
  return r;
}

// Guarded swizzled row load: -inf outside [0, Wn). Speculative address is
// masked in-bounds so a hoisted ds_load can never go out of range.
__device__ __forceinline__ v4f ld_w(const v4f* __restrict__ s, int w) {
  v4f v = s[wslot(w & (Wn - 1))];
  return (w >= 0 && w < Wn) ? v : neg_inf4();
}

// Horizontal sliding window-5 max along W, full row width, SAME padding with
// -inf. Each task scans a 16-wide run of one (row, c4) in registers:
// ~1.25 LDS reads per output element instead of 5.
__device__ __forceinline__ void hscan5(const v4f* __restrict__ src,
                                       v4f* __restrict__ dst, int nrows,
                                       int tid) {
  const int ntask = nrows * 16;  // nrows x 4 c4 x 4 w-runs
  for (int task = tid; task < ntask; task += NTHREADS) {
    const int c4 = task & 3;
    const int wrun = (task >> 2) & 3;
    const int row = task >> 4;
    const v4f* s = src + row * ROWF4 + c4;
    v4f* d = dst + row * ROWF4 + c4;
    const int w0 = wrun * 16;
    v4f a = ld_w(s, w0 - 2);
    v4f b = ld_w(s, w0 - 1);
    v4f c = ld_w(s, w0);
    v4f dd = ld_w(s, w0 + 1);
#pragma unroll
    for (int j = 0; j < 16; ++j) {
      const int w = w0 + j;
      v4f e = ld_w(s, w + 2);
      v4f m = vmax4(vmax4(vmax4(a, b), vmax4(c, dd)), e);
      d[wslot(w)] = m;
      a = b;
      b = c;
      c = dd;
      dd = e;
    }
  }
}

// Vertical window-5 max streaming rows through a 5-deep register ring:
// 1 LDS read per output element. Optionally writes the pooled row back to
// LDS and stores the TH center rows to global with a non-temporal hint.
__device__ __forceinline__ void vring5(const v4f* __restrict__ s,
                                       v4f* __restrict__ d, bool write_lds,
                                       int nout, int istore0,
                                       float* __restrict__ gcol, size_t hstride,
                                       int habs0) {
  v4f r0 = s[0 * ROWF4];
  v4f r1 = s[1 * ROWF4];
  v4f r2 = s[2 * ROWF4];
  v4f r3 = s[3 * ROWF4];
  for (int i = 0; i < nout; ++i) {
    v4f r4 = s[(i + 4) * ROWF4];
    v4f m = vmax4(vmax4(vmax4(r0, r1), vmax4(r2, r3)), r4);
    if (write_lds) d[i * ROWF4] = m;
    const int k = i - istore0;
    if (k >= 0 && k < TH) {
      __builtin_nontemporal_store(m,
                                  (v4f*)(gcol + (size_t)(habs0 + i) * hstride));
    }
    r0 = r1;
    r1 = r2;
    r2 = r3;
    r3 = r4;
  }
}

__global__ __launch_bounds__(NTHREADS) void spp_kernel(
    const float* __restrict__ x, float* __restrict__ out) {
  extern __shared__ v4f lds[];
  v4f* A = lds;        // raw -> p5 -> p9
  v4f* B = lds + BUF;  // h5 -> h9 -> h13

  const int tid = threadIdx.x;
  const int n = blockIdx.z;
  const int h0 = blockIdx.y * TH;
  const int cb = blockIdx.x * CC;

  const int w = tid >> 2;
  const int c4 = tid & 3;

  // Column base pointers for this thread's (w, c4) lane.
  const float* gxcol = x + ((size_t)n * Hn * Wn + w) * (size_t)Cn + cb + c4 * 4;
  float* gocol = out + ((size_t)n * Hn * Wn + w) * (size_t)OCn + cb + c4 * 4;
  const size_t ghstride_in = (size_t)Wn * Cn;    // floats per h row (input)
  const size_t ghstride_out = (size_t)Wn * OCn;  // floats per h row (output)

  // ---- Phase 0: raw tile (rows h0-6 .. h0+13) -> A; pass-through copy ----
  {
    v4f* lcol = A + wslot(w) + c4;
    for (int row = 0; row < RRAW; ++row) {
      const int habs = h0 - 6 + row;
      v4f* lp = lcol + row * ROWF4;
      if (habs >= 0 && habs < Hn) {
        const float* gp = gxcol + (size_t)habs * ghstride_in;
#if SPP_USE_ASYNC
        // gfx1250 async copy: LDS dest addr (low 32 bits of generic shared
        // pointer) in a VGPR, 64-bit global addr in a VGPR pair.
        unsigned laddr = (unsigned)(uintptr_t)(const void*)lp;
        unsigned long long gaddr = (unsigned long long)(uintptr_t)gp;
        asm volatile("global_load_async_to_lds_b128 %0, %1, off" ::"v"(laddr),
                     "v"(gaddr)
                     : "memory");
        if (habs >= h0 && habs < h0 + TH) {
          v4f v = *(const v4f*)gp;  // L2 hit; needed for channel-0 passthrough
          __builtin_nontemporal_store(
              v, (v4f*)(gocol + 0 * Cn + (size_t)habs * ghstride_out));
        }
#else
        v4f v = *(const v4f*)gp;
        *lp = v;
        if (habs >= h0 && habs < h0 + TH) {
          __builtin_nontemporal_store(
              v, (v4f*)(gocol + 0 * Cn + (size_t)habs * ghstride_out));
        }
#endif
      } else {
        *lp = neg_inf4();  // SAME padding with -inf
      }
    }
  }
#if SPP_USE_ASYNC
  asm volatile("s_wait_asynccnt 0" ::: "memory");
#endif
  __syncthreads();

  // ---- Level 1: h5 = hmax5(raw) : A(20) -> B(20) ----
  hscan5(A, B, RRAW, tid);
  __syncthreads();

  // p5 = vmax5(h5) : B(20) -> A(16 rows, abs h0-4..h0+11); store mp5 center.
  {
    const v4f* sB = B + wslot(w) + c4;
    v4f* dA = A + wslot(w) + c4;
    vring5(sB, dA, true, 16, 4, gocol + 1 * Cn, ghstride_out, h0 - 4);
  }
  __syncthreads();

  // ---- Level 2: h9 = hmax5(p5) : A(16) -> B(16) ----
  hscan5(A, B, 16, tid);
  __syncthreads();

  // p9 = vmax5(h9) : B(16) -> A(12 rows, abs h0-2..h0+9); store mp9 center.
  {
    const v4f* sB = B + wslot(w) + c4;
    v4f* dA = A + wslot(w) + c4;
    vring5(sB, dA, true, 12, 2, gocol + 2 * Cn, ghstride_out, h0 - 2);
  }
  __syncthreads();

  // ---- Level 3: h13 = hmax5(p9) : A(12) -> B(12) ----
  hscan5(A, B, 12, tid);
  __syncthreads();

  // p13 = vmax5(h13) : B(12) -> global only (8 rows, abs h0..h0+7).
  {
    const v4f* sB = B + wslot(w) + c4;
    v4f* dA = A + wslot(w) + c4;  // unused (write_lds=false)
    vring5(sB, dA, false, TH, 0, gocol + 3 * Cn, ghstride_out, h0);
  }
}

extern "C" void kernel_launch(void* const* d_in, const int* in_sizes, int n_in,
                              void* d_out, int out_size, void* d_ws,
                              size_t ws_size, hipStream_t stream) {
  (void)n_in;
  (void)out_size;
  (void)d_ws;
  (void)ws_size;
  const float* x = (const float*)d_in[0];
  float* out = (float*)d_out;

  const int N = in_sizes[0] / (Hn * Wn * Cn);  // 16 for the reference shapes
  const size_t shmem = (size_t)2 * BUF * sizeof(v4f);  // 163,840 B: 2/WGP

  // Allow >64KB dynamic LDS (no-op where not required); deterministic.
  (void)hipFuncSetAttribute(reinterpret_cast<const void*>(spp_kernel),
                            hipFuncAttributeMaxDynamicSharedMemorySize,
                            (int)shmem);

  dim3 grid(Cn / CC, Hn / TH, N);  // 32 x 8 x 16 = 4096 blocks
  spp_kernel<<<grid, NTHREADS, shmem, stream>>>(x, out);
}